// LaserCategoryEngine_47098611368144
// MI455X (gfx1250) — compile-verified
//
#include <hip/hip_runtime.h>
#include <math.h>

#define N_OBJ   32
#define M_PER   4
#define CPO     256          // cells per object
#define HID     1024
#define OM      128          // N_OBJ * M_PER
#define DECAY   0.81450625f  // 0.95^4

typedef float v2f __attribute__((ext_vector_type(2)));
typedef float v8f __attribute__((ext_vector_type(8)));

// ---------------------------------------------------------------------------
// K1: pump = sigmoid(x @ pump_W.T + b); obj_ex = clip(0.95*oe + 0.05*pump);
//     mask = obj_ex > 0.5.  Also zero the scalar accumulators.
// ---------------------------------------------------------------------------
__global__ void pump_kernel(const float* __restrict__ x,
                            const float* __restrict__ pW,
                            const float* __restrict__ pb,
                            const float* __restrict__ oex_in,
                            float* __restrict__ obj_ex,
                            float* __restrict__ maskf,
                            float* __restrict__ scal) {
  const int o = blockIdx.x;
  __shared__ float red[256];
  float s = 0.f;
  for (int i = threadIdx.x; i < HID; i += 256) s += x[i] * pW[o * HID + i];
  red[threadIdx.x] = s; __syncthreads();
  for (int st = 128; st > 0; st >>= 1) {
    if (threadIdx.x < st) red[threadIdx.x] += red[threadIdx.x + st];
    __syncthreads();
  }
  if (threadIdx.x == 0) {
    float p = 1.f / (1.f + expf(-(red[0] + pb[o])));
    float e = oex_in[o] * 0.95f + p * 0.05f;
    e = fminf(fmaxf(e, 0.f), 1.f);
    obj_ex[o] = e;
    maskf[o] = (e > 0.5f) ? 1.f : 0.f;
    if (o == 0) { scal[0] = 0.f; scal[1] = 0.f; scal[2] = 0.f; }
  }
}

// ---------------------------------------------------------------------------
// K2: src_mean[o][h] = mean over 256 cells (complex). grid (N_OBJ);
// thread t owns h = 4t..4t+3, streamed with b128 loads.
// ---------------------------------------------------------------------------
__global__ void src_mean_kernel(const float* __restrict__ cre,
                                const float* __restrict__ cim,
                                float* __restrict__ sre,
                                float* __restrict__ sim) {
  const int o  = blockIdx.x;
  const int h4 = threadIdx.x * 4;
  const float* pr = cre + (size_t)o * CPO * HID + h4;
  const float* pi = cim + (size_t)o * CPO * HID + h4;
  float4 ar = {0.f, 0.f, 0.f, 0.f}, ai = {0.f, 0.f, 0.f, 0.f};
  for (int c = 0; c < CPO; ++c) {
    const float4 r = *reinterpret_cast<const float4*>(pr + (size_t)c * HID);
    const float4 m = *reinterpret_cast<const float4*>(pi + (size_t)c * HID);
    ar.x += r.x; ar.y += r.y; ar.z += r.z; ar.w += r.w;
    ai.x += m.x; ai.y += m.y; ai.z += m.z; ai.w += m.w;
  }
  const float inv = 1.f / CPO;
  ar.x *= inv; ar.y *= inv; ar.z *= inv; ar.w *= inv;
  ai.x *= inv; ai.y *= inv; ai.z *= inv; ai.w *= inv;
  *reinterpret_cast<float4*>(sre + o * HID + h4) = ar;
  *reinterpret_cast<float4*>(sim + o * HID + h4) = ai;
}

// ---------------------------------------------------------------------------
// K3/K5: batched complex GEMV via V_WMMA_F32_16X16X4_F32.
//   out[om][:] = W[om] @ (vre[o], vim[o]),  o = om>>2.
// grid (HID/16 row tiles, OM); 256 threads = 8 waves; wave w owns K-range
// [w*128, w*128+128).
//
// One b128 load per lane feeds TWO wmma steps by permuting the K mapping
// (valid since the dot product is order-agnostic over K, and A and B use the
// same permutation):
//   pair p, kb = kbase+8p:
//     step0 K-slots: low half-lanes {kb,kb+1},  high half-lanes {kb+4,kb+5}
//     step1 K-slots: low half-lanes {kb+2,kb+3},high half-lanes {kb+6,kb+7}
// B columns 0/1 carry the re/im vector chunk (staged in LDS by
// GLOBAL_LOAD_ASYNC_TO_LDS_B128, waited with s_wait_asynccnt).
// ---------------------------------------------------------------------------
__global__ void gemv_wmma_kernel(const float* __restrict__ W,
                                 const float* __restrict__ vre,
                                 const float* __restrict__ vim,
                                 float* __restrict__ out_re,
                                 float* __restrict__ out_im) {
  const int om   = blockIdx.y;         // 0..127
  const int o    = om >> 2;
  const int r0   = blockIdx.x * 16;    // row tile base
  const int tid  = threadIdx.x;
  const int wave = tid >> 5;           // wave32
  const int lane = tid & 31;

  __shared__ __align__(16) float sv[2][HID];   // staged vector (re, im)
  __shared__ float red[8][256];                // per-wave partial C matrices

  // ---- async DMA stage: 256 threads x b128 covers one 1024-float vector ----
  {
    const unsigned goff  = (unsigned)(tid * 16);                  // byte offset
    const unsigned ldsa0 = (unsigned)(uintptr_t)(&sv[0][tid * 4]);
    const unsigned ldsa1 = (unsigned)(uintptr_t)(&sv[1][tid * 4]);
    const float* gre = vre + (size_t)o * HID;
    const float* gim = vim + (size_t)o * HID;
    asm volatile("global_load_async_to_lds_b128 %0, %1, %2"
                 :: "v"(ldsa0), "v"(goff), "s"(gre) : "memory");
    asm volatile("global_load_async_to_lds_b128 %0, %1, %2"
                 :: "v"(ldsa1), "v"(goff), "s"(gim) : "memory");
    asm volatile("s_wait_asynccnt 0x0" ::: "memory");
  }
  __syncthreads();

  const int   kbase = wave * 128;
  const int   rowA  = r0 + (lane & 15);
  const int   half  = lane >> 4;
  const int   col   = lane & 15;       // B column held by this lane
  const float* wrow = W + ((size_t)om * HID + rowA) * HID;

  __builtin_prefetch(wrow + kbase + 64, 0, 1);   // global_prefetch_b8

  v8f c = {0.f, 0.f, 0.f, 0.f, 0.f, 0.f, 0.f, 0.f};
#pragma unroll
  for (int p = 0; p < 16; ++p) {
    const int kb = kbase + p * 8;
    const float4 d = reinterpret_cast<const float4*>(wrow + kb)[half];
    float4 e = {0.f, 0.f, 0.f, 0.f};
    if (col < 2)                        // col 0 = re, col 1 = im
      e = reinterpret_cast<const float4*>(&sv[col][kb])[half];
    v2f a0; a0.x = d.x; a0.y = d.y;
    v2f b0; b0.x = e.x; b0.y = e.y;
    c = __builtin_amdgcn_wmma_f32_16x16x4_f32(
        false, a0, false, b0, (short)0, c, false, false);
    v2f a1; a1.x = d.z; a1.y = d.w;
    v2f b1; b1.x = e.z; b1.y = e.w;
    c = __builtin_amdgcn_wmma_f32_16x16x4_f32(
        false, a1, false, b1, (short)0, c, false, false);
  }

#pragma unroll
  for (int v = 0; v < 8; ++v) red[wave][lane * 8 + v] = c[v];
  __syncthreads();

  float sum = 0.f;
#pragma unroll
  for (int w2 = 0; w2 < 8; ++w2) sum += red[w2][tid];
  red[0][tid] = sum;                    // each thread owns its own slot
  __syncthreads();

  if (tid < 16) {
    const int r = tid;
    float re, im;
    if (r < 8) { re = red[0][0 * 8 + r];        im = red[0][1 * 8 + r]; }
    else       { re = red[0][16 * 8 + (r - 8)]; im = red[0][17 * 8 + (r - 8)]; }
    out_re[(size_t)om * HID + r0 + r] = re;
    out_im[(size_t)om * HID + r0 + r] = im;
  }
}

// ---------------------------------------------------------------------------
// K4: contrib via the ordered-scatter coefficients; obj_means.
// grid (HID/256, N_OBJ as target t)
// ---------------------------------------------------------------------------
__global__ void contrib_objm_kernel(const float* __restrict__ Tre,
                                    const float* __restrict__ Tim,
                                    const float* __restrict__ sre,
                                    const float* __restrict__ sim,
                                    float* __restrict__ ctr_re,
                                    float* __restrict__ ctr_im,
                                    float* __restrict__ omr,
                                    float* __restrict__ omi) {
  const int t = blockIdx.y;
  const int h = blockIdx.x * 256 + threadIdx.x;
  int os[4];
#pragma unroll
  for (int m = 0; m < 4; ++m) os[m] = (t - 1 - m) & 31;
  // coef by ascending-source rank p: 0.05 * 0.95^(3-p)
  const float ctab[4] = {0.04286875f, 0.045125f, 0.0475f, 0.05f};
  float ar = 0.f, ai = 0.f;
#pragma unroll
  for (int m = 0; m < 4; ++m) {
    const int oo = os[m];
    int p = 0;
#pragma unroll
    for (int j = 0; j < 4; ++j) p += (os[j] < oo) ? 1 : 0;
    const float cf = ctab[p];
    const size_t row = (size_t)(oo * 4 + m) * HID + h;
    ar += cf * Tre[row];
    ai += cf * Tim[row];
  }
  ctr_re[t * HID + h] = ar;
  ctr_im[t * HID + h] = ai;
  omr[t * HID + h] = DECAY * sre[t * HID + h] + ar;
  omi[t * HID + h] = DECAY * sim[t * HID + h] + ai;
}

// ---------------------------------------------------------------------------
// K6: colimit = mean(morphed), limit = mean(obj_means);
//     accumulate sum_h |limit-colimit|^2 into scal[0]. grid (HID/256)
// ---------------------------------------------------------------------------
__global__ void colimit_kernel(const float* __restrict__ Mre,
                               const float* __restrict__ Mim,
                               const float* __restrict__ omr,
                               const float* __restrict__ omi,
                               float* __restrict__ scal) {
  const int h = blockIdx.x * 256 + threadIdx.x;
  float cr = 0.f, ci = 0.f;
  for (int r = 0; r < OM; ++r) {
    cr += Mre[(size_t)r * HID + h];
    ci += Mim[(size_t)r * HID + h];
  }
  cr *= (1.f / OM); ci *= (1.f / OM);
  float lr = 0.f, li = 0.f;
  for (int oo = 0; oo < N_OBJ; ++oo) {
    lr += omr[oo * HID + h];
    li += omi[oo * HID + h];
  }
  lr *= (1.f / N_OBJ); li *= (1.f / N_OBJ);
  const float dr = lr - cr, di = li - ci;
  float d = dr * dr + di * di;
  __shared__ float red[256];
  red[threadIdx.x] = d; __syncthreads();
  for (int st = 128; st > 0; st >>= 1) {
    if (threadIdx.x < st) red[threadIdx.x] += red[threadIdx.x + st];
    __syncthreads();
  }
  if (threadIdx.x == 0) atomicAdd(&scal[0], red[0]);
}

// ---------------------------------------------------------------------------
// K7a: per-object mean of post-lasing cells (for las_mean). grid (N_OBJ);
// thread t owns h = 4t..4t+3, b128 streams.
// ---------------------------------------------------------------------------
__global__ void objpost_kernel(const float* __restrict__ cre,
                               const float* __restrict__ cim,
                               const float* __restrict__ ctr_re,
                               const float* __restrict__ ctr_im,
                               const float* __restrict__ maskf,
                               const float* __restrict__ cavr,
                               const float* __restrict__ cavi,
                               float* __restrict__ opr,
                               float* __restrict__ opi) {
  const int o  = blockIdx.x;
  const int h4 = threadIdx.x * 4;
  const float4 cr0 = *reinterpret_cast<const float4*>(ctr_re + o * HID + h4);
  const float4 ci0 = *reinterpret_cast<const float4*>(ctr_im + o * HID + h4);
  const bool msk = maskf[o] > 0.5f;
  float ca[4];
#pragma unroll
  for (int j = 0; j < 4; ++j) ca[j] = atan2f(cavi[h4 + j], cavr[h4 + j]);
  const float* pr = cre + (size_t)o * CPO * HID + h4;
  const float* pi = cim + (size_t)o * CPO * HID + h4;
  float sr[4] = {0.f, 0.f, 0.f, 0.f}, si[4] = {0.f, 0.f, 0.f, 0.f};
  for (int cidx = 0; cidx < CPO; ++cidx) {
    const float4 r = *reinterpret_cast<const float4*>(pr + (size_t)cidx * HID);
    const float4 m = *reinterpret_cast<const float4*>(pi + (size_t)cidx * HID);
    float nr[4] = {DECAY * r.x + cr0.x, DECAY * r.y + cr0.y,
                   DECAY * r.z + cr0.z, DECAY * r.w + cr0.w};
    float ni[4] = {DECAY * m.x + ci0.x, DECAY * m.y + ci0.y,
                   DECAY * m.z + ci0.z, DECAY * m.w + ci0.w};
    if (msk) {
#pragma unroll
      for (int j = 0; j < 4; ++j) {
        const float ang = 0.3f * ca[j] + 0.7f * atan2f(ni[j], nr[j]);
        const float amp = sqrtf(nr[j] * nr[j] + ni[j] * ni[j]);
        float sn, cs; sincosf(ang, &sn, &cs);
        nr[j] = amp * cs; ni[j] = amp * sn;
      }
    }
#pragma unroll
    for (int j = 0; j < 4; ++j) { sr[j] += nr[j]; si[j] += ni[j]; }
  }
  float4 outr, outi;
  outr.x = sr[0] * (1.f / CPO); outr.y = sr[1] * (1.f / CPO);
  outr.z = sr[2] * (1.f / CPO); outr.w = sr[3] * (1.f / CPO);
  outi.x = si[0] * (1.f / CPO); outi.y = si[1] * (1.f / CPO);
  outi.z = si[2] * (1.f / CPO); outi.w = si[3] * (1.f / CPO);
  *reinterpret_cast<float4*>(opr + o * HID + h4) = outr;
  *reinterpret_cast<float4*>(opi + o * HID + h4) = outi;
}

// ---------------------------------------------------------------------------
// K7b: per-cell amplitude normalization + variance stats.
// Lasing preserves |new|, so no trig needed. grid (N_CELLS)
// ---------------------------------------------------------------------------
__global__ void stats_kernel(const float* __restrict__ cre,
                             const float* __restrict__ cim,
                             const float* __restrict__ ctr_re,
                             const float* __restrict__ ctr_im,
                             float* __restrict__ scal) {
  const int cell = blockIdx.x;
  const int o = cell >> 8;
  const int t = threadIdx.x;
  const int h4 = t * 4;
  const float4 r  = *reinterpret_cast<const float4*>(cre + (size_t)cell * HID + h4);
  const float4 m  = *reinterpret_cast<const float4*>(cim + (size_t)cell * HID + h4);
  const float4 c0 = *reinterpret_cast<const float4*>(ctr_re + o * HID + h4);
  const float4 c1 = *reinterpret_cast<const float4*>(ctr_im + o * HID + h4);
  float nr[4] = {DECAY * r.x + c0.x, DECAY * r.y + c0.y,
                 DECAY * r.z + c0.z, DECAY * r.w + c0.w};
  float ni[4] = {DECAY * m.x + c1.x, DECAY * m.y + c1.y,
                 DECAY * m.z + c1.z, DECAY * m.w + c1.w};
  float amps[4];
  float mymax = 0.f;
#pragma unroll
  for (int j = 0; j < 4; ++j) {
    const float a = sqrtf(nr[j] * nr[j] + ni[j] * ni[j]);
    amps[j] = a;
    mymax = fmaxf(mymax, a);
  }
  __shared__ float red[256];
  red[t] = mymax; __syncthreads();
  for (int st = 128; st > 0; st >>= 1) {
    if (t < st) red[t] = fmaxf(red[t], red[t + st]);
    __syncthreads();
  }
  const float rmax = red[0] + 1e-8f;
  __syncthreads();
  float s1 = 0.f, s2 = 0.f;
#pragma unroll
  for (int j = 0; j < 4; ++j) {
    const float a = amps[j] / rmax;
    s1 += a; s2 += a * a;
  }
  red[t] = s1; __syncthreads();
  for (int st = 128; st > 0; st >>= 1) {
    if (t < st) red[t] += red[t + st];
    __syncthreads();
  }
  const float bs1 = red[0];
  __syncthreads();
  red[t] = s2; __syncthreads();
  for (int st = 128; st > 0; st >>= 1) {
    if (t < st) red[t] += red[t + st];
    __syncthreads();
  }
  if (t == 0) {
    atomicAdd(&scal[1], bs1);
    atomicAdd(&scal[2], red[0]);
  }
}

// ---------------------------------------------------------------------------
// K8a: cavity update -> cav2[2048] = [cav.re', cav.im']. grid (2048/256)
// ---------------------------------------------------------------------------
__global__ void cavity_kernel(const float* __restrict__ maskf,
                              const float* __restrict__ opr,
                              const float* __restrict__ opi,
                              const float* __restrict__ cavr,
                              const float* __restrict__ cavi,
                              float* __restrict__ cav2) {
  const int idx = blockIdx.x * 256 + threadIdx.x;  // 0..2047
  float nl = 0.f;
  for (int o = 0; o < N_OBJ; ++o) nl += (maskf[o] > 0.5f) ? 1.f : 0.f;
  const int h = idx & (HID - 1);
  const bool isim = idx >= HID;
  const float* op = isim ? opi : opr;
  float s = 0.f;
  for (int o = 0; o < N_OBJ; ++o)
    if (maskf[o] > 0.5f) s += op[o * HID + h];
  const float lm = s / fmaxf(nl, 1.f);
  const float cv = isim ? cavi[h] : cavr[h];
  cav2[idx] = (nl > 0.f) ? (0.8f * cv + 0.2f * lm) : cv;
}

// ---------------------------------------------------------------------------
// K8b: pred = cav2 @ dec_W.T + dec_b; block 0 also emits tension. grid (1024)
// ---------------------------------------------------------------------------
__global__ void pred_kernel(const float* __restrict__ cav2,
                            const float* __restrict__ dW,
                            const float* __restrict__ db,
                            const float* __restrict__ scal,
                            float* __restrict__ out) {
  const int j = blockIdx.x;
  __shared__ float red[256];
  const float* w = dW + (size_t)j * (2 * HID);
  float s = 0.f;
  for (int k = threadIdx.x; k < 2 * HID; k += 256) s += cav2[k] * w[k];
  red[threadIdx.x] = s; __syncthreads();
  for (int st = 128; st > 0; st >>= 1) {
    if (threadIdx.x < st) red[threadIdx.x] += red[threadIdx.x + st];
    __syncthreads();
  }
  if (threadIdx.x == 0) {
    out[j] = red[0] + db[j];
    if (j == 0) {
      const float N = 8388608.f;                 // 8192*1024
      const float mean = scal[1] / N;
      const float var = (scal[2] - scal[1] * mean) / (N - 1.f);
      out[HID] = scal[0] * (1.f / (float)HID) + var;
    }
  }
}

// ---------------------------------------------------------------------------
extern "C" void kernel_launch(void* const* d_in, const int* in_sizes, int n_in,
                              void* d_out, int out_size, void* d_ws, size_t ws_size,
                              hipStream_t stream) {
  (void)in_sizes; (void)n_in; (void)out_size; (void)ws_size;
  const float* x    = (const float*)d_in[0];
  const float* cre  = (const float*)d_in[1];
  const float* cim  = (const float*)d_in[2];
  const float* cavr = (const float*)d_in[3];
  const float* cavi = (const float*)d_in[4];
  const float* oex  = (const float*)d_in[5];
  const float* W    = (const float*)d_in[6];
  const float* pW   = (const float*)d_in[7];
  const float* pb   = (const float*)d_in[8];
  const float* dW   = (const float*)d_in[9];
  const float* db   = (const float*)d_in[10];
  float* out = (float*)d_out;
  float* ws  = (float*)d_ws;

  // workspace layout (float offsets)
  constexpr size_t OFF_OBJEX = 0;                        // 32
  constexpr size_t OFF_MASK  = 32;                       // 32
  constexpr size_t OFF_SCAL  = 64;                       // 8
  constexpr size_t OFF_SRE   = 128;                      // 32*1024
  constexpr size_t OFF_SIM   = OFF_SRE + 32768;
  constexpr size_t OFF_TRE   = OFF_SIM + 32768;          // 128*1024
  constexpr size_t OFF_TIM   = OFF_TRE + 131072;
  constexpr size_t OFF_CRE   = OFF_TIM + 131072;         // contrib
  constexpr size_t OFF_CIM   = OFF_CRE + 32768;
  constexpr size_t OFF_OMR   = OFF_CIM + 32768;          // obj_means
  constexpr size_t OFF_OMI   = OFF_OMR + 32768;
  constexpr size_t OFF_MRE   = OFF_OMI + 32768;          // morphed
  constexpr size_t OFF_MIM   = OFF_MRE + 131072;
  constexpr size_t OFF_OPR   = OFF_MIM + 131072;         // objpost
  constexpr size_t OFF_OPI   = OFF_OPR + 32768;
  constexpr size_t OFF_CAV2  = OFF_OPI + 32768;          // 2048

  pump_kernel<<<N_OBJ, 256, 0, stream>>>(x, pW, pb, oex,
      ws + OFF_OBJEX, ws + OFF_MASK, ws + OFF_SCAL);

  src_mean_kernel<<<N_OBJ, 256, 0, stream>>>(
      cre, cim, ws + OFF_SRE, ws + OFF_SIM);

  // pass 1: T = W @ src_mean
  gemv_wmma_kernel<<<dim3(HID / 16, OM), 256, 0, stream>>>(
      W, ws + OFF_SRE, ws + OFF_SIM, ws + OFF_TRE, ws + OFF_TIM);

  contrib_objm_kernel<<<dim3(HID / 256, N_OBJ), 256, 0, stream>>>(
      ws + OFF_TRE, ws + OFF_TIM, ws + OFF_SRE, ws + OFF_SIM,
      ws + OFF_CRE, ws + OFF_CIM, ws + OFF_OMR, ws + OFF_OMI);

  // pass 2: morphed = W @ obj_means
  gemv_wmma_kernel<<<dim3(HID / 16, OM), 256, 0, stream>>>(
      W, ws + OFF_OMR, ws + OFF_OMI, ws + OFF_MRE, ws + OFF_MIM);

  colimit_kernel<<<HID / 256, 256, 0, stream>>>(
      ws + OFF_MRE, ws + OFF_MIM, ws + OFF_OMR, ws + OFF_OMI, ws + OFF_SCAL);

  objpost_kernel<<<N_OBJ, 256, 0, stream>>>(
      cre, cim, ws + OFF_CRE, ws + OFF_CIM, ws + OFF_MASK,
      cavr, cavi, ws + OFF_OPR, ws + OFF_OPI);

  stats_kernel<<<8192, 256, 0, stream>>>(
      cre, cim, ws + OFF_CRE, ws + OFF_CIM, ws + OFF_SCAL);

  cavity_kernel<<<2 * HID / 256, 256, 0, stream>>>(
      ws + OFF_MASK, ws + OFF_OPR, ws + OFF_OPI, cavr, cavi, ws + OFF_CAV2);

  pred_kernel<<<HID, 256, 0, stream>>>(
      ws + OFF_CAV2, dW, db, ws + OFF_SCAL, out);
}